// SimpleDistributionShiftTracer_38268158608011
// MI455X (gfx1250) — compile-verified
//
#include <hip/hip_runtime.h>
#include <hip/hip_bf16.h>
#include <stdint.h>

#define T_LEN 4096
#define C_LEN 128
#define B_LEN 32
#define TT    64              // t-positions per tile
#define CT    64              // channels per tile (= blockDim.x)
#define HALO  4               // output t depends on x[t-4 .. t+4]
#define COLS  (TT + 2*HALO)   // 72 staged columns per row
#define PITCH 76              // LDS row pitch (floats): 16B-aligned rows, conflict-light

// ---- CDNA5 async tensor-data-movement path (gfx1250) ----
__device__ __forceinline__ void async_load_b128(uint32_t lds_byte_off, const float* gptr) {
  asm volatile("global_load_async_to_lds_b128 %0, %1, off"
               :: "v"(lds_byte_off), "v"(gptr) : "memory");
}
__device__ __forceinline__ void async_store_b128(float* gptr, uint32_t lds_byte_off) {
  asm volatile("global_store_async_from_lds_b128 %0, %1, off"
               :: "v"(gptr), "v"(lds_byte_off) : "memory");
}
__device__ __forceinline__ void wait_async0() {
  asm volatile("s_wait_asynccnt 0x0" ::: "memory");
}

// Kernel 1: per-(b,c) row mean over T=4096.  4096 blocks x 256 threads.
__global__ void row_mean_kernel(const float* __restrict__ x, float* __restrict__ mean) {
  const int row = blockIdx.x;                       // 0..4095 = b*C + c
  const float4* p = (const float4*)(x + (size_t)row * T_LEN);
  float s = 0.f;
  for (int i = threadIdx.x; i < T_LEN / 4; i += 256) {
    float4 v = p[i];
    s += v.x + v.y + v.z + v.w;
  }
  #pragma unroll
  for (int off = 16; off > 0; off >>= 1) s += __shfl_down(s, off, 32);
  __shared__ float red[8];
  if ((threadIdx.x & 31) == 0) red[threadIdx.x >> 5] = s;
  __syncthreads();
  if (threadIdx.x == 0) {
    float t = 0.f;
    #pragma unroll
    for (int w = 0; w < 8; w++) t += red[w];
    mean[row] = t * (1.f / (float)T_LEN);
  }
}

// Kernel 2: fused rolling stats + depthwise conv + leaky + residual + transpose.
__launch_bounds__(CT)
__global__ void fused_kernel(const float* __restrict__ x,
                             const float* __restrict__ wgt,   // (C,1,6,5) flat: c*30 + f*5 + k
                             const float* __restrict__ bias,  // (C)
                             const float* __restrict__ mean,  // (B*C) from d_ws
                             float* __restrict__ out) {       // (B,T,C)
  __shared__ float Xs[CT * PITCH];   // staged input rows
  __shared__ float Tb[TT * CT];      // transposed output tile [t][c]

  const int tid   = threadIdx.x;          // = local channel
  const int ttile = blockIdx.x;           // 0..63
  const int ctile = blockIdx.y;           // 0..1
  const int b     = blockIdx.z;           // 0..31
  const int t0    = ttile * TT;
  const int cl    = tid;
  const int cg    = ctile * CT + cl;      // global channel
  const bool edge = (ttile == 0) || (ttile == (T_LEN / TT - 1));

  const size_t rowbase = ((size_t)(b * C_LEN + cg)) * T_LEN;

  // ---- stage x[t0-4 .. t0+TT+4) per channel row into LDS ----
  if (!edge) {
    // interior tile: fully in-range, 16B aligned -> async DMA to LDS
    const float* gp   = x + rowbase + (t0 - HALO);
    uint32_t     lb   = (uint32_t)(uintptr_t)(&Xs[cl * PITCH]);
    #pragma unroll
    for (int q = 0; q < COLS / 4; q++)            // 18 x B128 per row
      async_load_b128(lb + q * 16, gp + q * 4);
    wait_async0();                                 // each wave reads only rows it loaded
  } else {
    for (int idx = tid; idx < CT * COLS; idx += CT) {
      int c = idx / COLS, j = idx - c * COLS;
      int tg = t0 - HALO + j;
      float v = 0.f;                               // zero-pad (matches avg-pool & conv pad)
      if ((unsigned)tg < (unsigned)T_LEN)
        v = x[((size_t)(b * C_LEN + ctile * CT + c)) * T_LEN + tg];
      Xs[c * PITCH + j] = v;
    }
    __syncthreads();
  }

  // ---- per-channel constants in registers ----
  float wk[30];
  {
    const float* wp = wgt + cg * 30;
    #pragma unroll
    for (int i = 0; i < 30; i++) wk[i] = wp[i];
  }
  const float bia  = bias[cg];
  const float mval = mean[b * C_LEN + cg];

  const float* Xrow = &Xs[cl * PITCH];
  float wa[5], wb[5], wc[5], acc[5];               // raw / max-masked / min-masked / conv ring
  #pragma unroll
  for (int a = 0; a < 5; a++) acc[a] = 0.f;
  #pragma unroll
  for (int a = 0; a < 4; a++) {
    float v = Xrow[a];
    float vb = v, vc = v;
    if (edge) {
      int tg = t0 - HALO + a;
      bool ok = (unsigned)tg < (unsigned)T_LEN;
      vb = ok ? v : -3.402823466e38f;              // -inf pad for max
      vc = ok ? v :  3.402823466e38f;              // +inf pad for min
    }
    wa[a] = v; wb[a] = vb; wc[a] = vc;
  }

  for (int i = 0; i < TT + HALO; i++) {            // 68 feature positions: t = t0-2+i
    float v = Xrow[i + 4];
    float vb = v, vc = v;
    if (edge) {
      int tg = t0 + i;
      bool ok = (unsigned)tg < (unsigned)T_LEN;
      vb = ok ? v : -3.402823466e38f;
      vc = ok ? v :  3.402823466e38f;
    }
    wa[4] = v; wb[4] = vb; wc[4] = vc;

    float sum = wa[0] + wa[1] + wa[2] + wa[3] + wa[4];
    float ssq = wa[0]*wa[0] + wa[1]*wa[1] + wa[2]*wa[2] + wa[3]*wa[3] + wa[4]*wa[4];
    float avg = sum * 0.2f;
    float var = fmaxf(ssq * 0.2f - avg * avg, 1e-6f);
    float sd  = sqrtf(var);
    float xc  = wa[2];
    float d   = xc - avg;
    float tm  = d * d * d;
    float mx  = fmaxf(fmaxf(fmaxf(wb[0], wb[1]), fmaxf(wb[2], wb[3])), wb[4]);
    float mn  = fminf(fminf(fminf(wc[0], wc[1]), fminf(wc[2], wc[3])), wc[4]);

    float f0 = xc, f1 = mval, f2 = sd, f3 = tm, f4 = mx, f5 = mn;
    if (edge) {                                    // conv zero-pads the feature map itself
      int tp = t0 - 2 + i;
      if ((unsigned)tp >= (unsigned)T_LEN) { f0 = f1 = f2 = f3 = f4 = f5 = 0.f; }
    }

    // scatter this feature vector into the 5 pending conv accumulators
    #pragma unroll
    for (int a = 0; a < 5; a++) {
      const int k = 4 - a;                         // acc[0] = oldest output (tap k=4)
      float s = acc[a];
      s = fmaf(f0, wk[0*5 + k], s);
      s = fmaf(f1, wk[1*5 + k], s);
      s = fmaf(f2, wk[2*5 + k], s);
      s = fmaf(f3, wk[3*5 + k], s);
      s = fmaf(f4, wk[4*5 + k], s);
      s = fmaf(f5, wk[5*5 + k], s);
      acc[a] = s;
    }

    if (i >= HALO) {                               // output t_local = i-4 now complete
      float o = acc[0] + bia;
      o = (o > 0.f) ? o : 0.01f * o;               // leaky ReLU
      Tb[(i - HALO) * CT + cl] = wa[0] + o;        // residual x[t] + conv; transposed store
    }
    #pragma unroll
    for (int a = 0; a < 4; a++) { wa[a]=wa[a+1]; wb[a]=wb[a+1]; wc[a]=wc[a+1]; acc[a]=acc[a+1]; }
    acc[4] = 0.f;
  }

  __syncthreads();                                 // Tb complete across waves
  asm volatile("s_wait_dscnt 0x0" ::: "memory");

  // ---- stream transposed tile to (B,T,C): each t-row is 64 contiguous floats, rows C_LEN apart ----
  float*   obase = out + ((size_t)b * T_LEN + t0) * C_LEN + ctile * CT;
  uint32_t tb0   = (uint32_t)(uintptr_t)(&Tb[0]);
  #pragma unroll
  for (int it = 0; it < (TT * CT) / (4 * CT); it++) {   // 16 x B128 per thread
    int idx4 = tid + it * CT;                      // 0..1023 float4s
    int trow = idx4 >> 4;
    int c4   = idx4 & 15;
    async_store_b128(obase + (size_t)trow * C_LEN + c4 * 4,
                     tb0 + (uint32_t)((trow * CT + c4 * 4) * 4));
  }
  wait_async0();                                   // drain before S_ENDPGM (also implicit)
}

extern "C" void kernel_launch(void* const* d_in, const int* in_sizes, int n_in,
                              void* d_out, int out_size, void* d_ws, size_t ws_size,
                              hipStream_t stream) {
  const float* x  = (const float*)d_in[0];   // (32,128,4096)
  const float* w  = (const float*)d_in[1];   // (128,1,6,5)
  const float* bi = (const float*)d_in[2];   // (128)
  float* out  = (float*)d_out;               // (32,4096,128)
  float* mean = (float*)d_ws;                // B*C = 4096 floats scratch

  row_mean_kernel<<<dim3(B_LEN * C_LEN), dim3(256), 0, stream>>>(x, mean);
  fused_kernel<<<dim3(T_LEN / TT, C_LEN / CT, B_LEN), dim3(CT), 0, stream>>>(x, w, bi, mean, out);
}